// RoutingLayer_13993003450623
// MI455X (gfx1250) — compile-verified
//
#include <hip/hip_runtime.h>
#include <hip/hip_bf16.h>
#include <cstdint>

typedef __attribute__((ext_vector_type(2))) float v2f;
typedef __attribute__((ext_vector_type(8))) float v8f;

#define B_DIM 32
#define I_DIM 512
#define O_DIM 64
#define DI_DIM 64
#define DO_DIM 64

// -----------------------------------------------------------------------------
// Kernel 1: priors[o,i,b,p] = sum_q W[o,i,p,q] * x[b,q,i]
// GEMM per (o,i): M=DO=64 (p), N=B=32 (b), K=DI=64 (q) via V_WMMA_F32_16X16X4_F32.
// grid = (64, 16): blockIdx.x = i-chunk of 8 (one i per wave), blockIdx.y = group of 4 o.
// block = 256 threads = 8 waves (wave32).
// W (512 MB) is streamed exactly once -> non-temporal loads so it does not
// sweep the 192 MB L2; priors stores stay temporal so K2 can hit them in L2.
// -----------------------------------------------------------------------------
template <typename T>
__global__ __launch_bounds__(256) void priors_kernel(
    const float* __restrict__ x,   // [B, DI, I]
    const float* __restrict__ W,   // [O, I, DO, DI]
    T* __restrict__ P)             // [O, I, B, DO]
{
    const int lane = threadIdx.x & 31;
    const int wave = threadIdx.x >> 5;
    const int i    = blockIdx.x * 8 + wave;
    const int ln   = lane & 15;
    const bool hi  = (lane >= 16);
    const int qoff = hi ? 2 : 0;

    // --- Load B fragments once per i (reused across the 4-o loop) ---------
    // B[k=q, n=b] = x[b, q, i].  Fragment (ks, nt): 2 floats, q = ks*4+qoff+{0,1},
    // column n = nt*16 + ln.  x is tiny (4 MB) and heavily reused: regular loads.
    v2f Bfrag[16][2];
#pragma unroll
    for (int ks = 0; ks < 16; ++ks) {
#pragma unroll
        for (int nt = 0; nt < 2; ++nt) {
            const int b  = nt * 16 + ln;
            const int q0 = ks * 4 + qoff;
            const float* xp = x + (size_t)b * (DI_DIM * I_DIM) + (size_t)q0 * I_DIM + i;
            v2f f;
            f.x = xp[0];
            f.y = xp[I_DIM];
            Bfrag[ks][nt] = f;
        }
    }

    for (int og = 0; og < 4; ++og) {
        const int o = blockIdx.y * 4 + og;
        const float* Wb = W + (size_t)(o * I_DIM + i) * (DO_DIM * DI_DIM);

        v8f acc[4][2];
#pragma unroll
        for (int mt = 0; mt < 4; ++mt)
#pragma unroll
            for (int nt = 0; nt < 2; ++nt)
#pragma unroll
                for (int e = 0; e < 8; ++e)
                    acc[mt][nt][e] = 0.0f;

#pragma unroll
        for (int ks = 0; ks < 16; ++ks) {
            // A[m=p, k=q] = W[o,i,p,q]; row m = mt*16 + ln, k = ks*4+qoff+{0,1}
            v2f A[4];
#pragma unroll
            for (int mt = 0; mt < 4; ++mt) {
                const float* ap = Wb + (size_t)(mt * 16 + ln) * DI_DIM + ks * 4 + qoff;
                A[mt] = __builtin_nontemporal_load((const v2f*)ap); // streamed once
            }
#pragma unroll
            for (int mt = 0; mt < 4; ++mt)
#pragma unroll
                for (int nt = 0; nt < 2; ++nt)
                    acc[mt][nt] = __builtin_amdgcn_wmma_f32_16x16x4_f32(
                        /*neg_a=*/false, A[mt],
                        /*neg_b=*/false, Bfrag[ks][nt],
                        /*c_mod=*/(short)0, acc[mt][nt],
                        /*reuse_a=*/false, /*reuse_b=*/false);
        }

        // --- Store D tiles: lane holds rows m = mt*16 + r + (hi?8:0), col b = nt*16+ln.
        // P layout [O,I,B,DO] -> 8 consecutive T per lane (temporal: K2 re-reads).
#pragma unroll
        for (int mt = 0; mt < 4; ++mt)
#pragma unroll
            for (int nt = 0; nt < 2; ++nt) {
                const int b = nt * 16 + ln;
                T* dst = P + (((size_t)(o * I_DIM + i)) * B_DIM + b) * DO_DIM
                           + mt * 16 + (hi ? 8 : 0);
#pragma unroll
                for (int e = 0; e < 8; ++e)
                    dst[e] = (T)acc[mt][nt][e];
            }
    }
}

// -----------------------------------------------------------------------------
// Kernel 2: dynamic routing per (b,o). priors slice [I=512, DO=64] in LDS (128KB).
// logits[i,p] = priors[i,p] * c[p] where c accumulates squashed outputs.
// block = 512 threads: p = t&63, group g = t>>6 handles i = g*64 .. g*64+63.
// priors reads are last-use -> non-temporal.
// -----------------------------------------------------------------------------
template <typename T>
__global__ __launch_bounds__(512) void routing_kernel(
    const T* __restrict__ P,       // [O, I, B, DO]
    float* __restrict__ out)       // [B, DO, O]
{
    extern __shared__ float smem[];
    float* Pr   = smem;                 // 32768 floats
    float* redA = smem + 32768;         // 512
    float* redB = redA + 512;           // 512
    float* sv   = redB + 512;           // 64

    const int t  = threadIdx.x;
    const int p  = t & 63;
    const int g  = t >> 6;
    const int bo = blockIdx.x;
    const int o  = bo & 63;
    const int b  = bo >> 6;

    // Load priors[b,o,:,:] -> LDS (coalesced: 64 threads cover one i-row of 64 floats)
    {
        const size_t base = ((size_t)o * I_DIM) * (B_DIM * DO_DIM) + (size_t)b * DO_DIM;
        for (int k = 0; k < 64; ++k) {
            const int lin = k * 512 + t;
            const int ii  = lin >> 6;
            const int pp  = lin & 63;
            Pr[lin] = (float)__builtin_nontemporal_load(
                          P + base + (size_t)ii * (B_DIM * DO_DIM) + pp);
        }
    }
    __syncthreads();

    float c_p   = 0.0f;
    float out_p = 0.0f;

    for (int it = 0; it < 3; ++it) {
        float s_p;
        if (it == 0) {
            // softmax(zeros) over i -> uniform 1/512: s = mean_i priors
            float ps = 0.0f;
            for (int k = 0; k < 64; ++k) ps += Pr[(g * 64 + k) * 64 + p];
            redA[g * 64 + p] = ps;
            __syncthreads();
            float tot = 0.0f;
            for (int g2 = 0; g2 < 8; ++g2) tot += redA[g2 * 64 + p];
            s_p = tot * (1.0f / 512.0f);
            __syncthreads();
        } else {
            // stable softmax over i of priors[i,p]*c_p, fused numerator
            float pm = -3.402823466e38f;
            for (int k = 0; k < 64; ++k)
                pm = fmaxf(pm, Pr[(g * 64 + k) * 64 + p] * c_p);
            redA[g * 64 + p] = pm;
            __syncthreads();
            float m = -3.402823466e38f;
            for (int g2 = 0; g2 < 8; ++g2) m = fmaxf(m, redA[g2 * 64 + p]);
            __syncthreads();
            float se = 0.0f, sn = 0.0f;
            for (int k = 0; k < 64; ++k) {
                const float pr = Pr[(g * 64 + k) * 64 + p];
                const float e  = __expf(pr * c_p - m);
                se += e;
                sn += e * pr;
            }
            redA[g * 64 + p] = se;
            redB[g * 64 + p] = sn;
            __syncthreads();
            float Z = 0.0f, S = 0.0f;
            for (int g2 = 0; g2 < 8; ++g2) {
                Z += redA[g2 * 64 + p];
                S += redB[g2 * 64 + p];
            }
            s_p = S / Z;
            __syncthreads();
        }

        // squash over p: sq = sum_p s_p^2
        if (g == 0) sv[p] = s_p;
        __syncthreads();
        float sq = 0.0f;
        for (int p2 = 0; p2 < 64; ++p2) {
            const float v = sv[p2];
            sq += v * v;
        }
        const float scale = sq / (1.0f + sq);
        out_p = scale * s_p * rsqrtf(sq);
        c_p += out_p;
        __syncthreads();
    }

    if (t < 64)
        out[(size_t)b * (DO_DIM * O_DIM) + (size_t)p * O_DIM + o] = out_p;
}

// -----------------------------------------------------------------------------
extern "C" void kernel_launch(void* const* d_in, const int* in_sizes, int n_in,
                              void* d_out, int out_size, void* d_ws, size_t ws_size,
                              hipStream_t stream)
{
    (void)in_sizes; (void)n_in; (void)out_size;
    const float* x = (const float*)d_in[0];        // [32, 64, 512] fp32
    const float* W = (const float*)d_in[1];        // [64, 512, 64, 64] fp32
    float* out = (float*)d_out;                    // [32, 64, 64] fp32

    const size_t PRI_ELEMS = (size_t)O_DIM * I_DIM * B_DIM * DO_DIM; // 64M
    const dim3 g1(64, 16), b1(256);
    const dim3 g2(2048), b2(512);
    const size_t lds2 = (size_t)(32768 + 512 + 512 + 64) * sizeof(float);

    if (ws_size >= PRI_ELEMS * sizeof(float)) {
        float* P = (float*)d_ws;
        priors_kernel<float><<<g1, b1, 0, stream>>>(x, W, P);
        routing_kernel<float><<<g2, b2, lds2, stream>>>(P, out);
    } else {
        // fp16-staged priors fallback (128 MB workspace, fully L2-resident);
        // compute stays fp32.
        _Float16* P = (_Float16*)d_ws;
        priors_kernel<_Float16><<<g1, b1, 0, stream>>>(x, W, P);
        routing_kernel<_Float16><<<g2, b2, lds2, stream>>>(P, out);
    }
}